// HarmonicOscillator_1090921693837
// MI455X (gfx1250) — compile-verified
//
#include <hip/hip_runtime.h>
#include <math.h>

// Problem constants (from the reference)
#define NB        16      // batch
#define NHARM     16      // harmonics
#define LFRAMES   256     // frame-rate length
#define SEG       960     // samples per frame interval (LW/LF)
#define LW        245760  // waveform samples
#define HEAD      480     // leading samples where interp pos clips to 0
#define TAILSTART 245280  // LW - 480: trailing samples where pos clips to 255

#define TILES_PER_WAVE   15     // 15 tiles x 16 samples = 240 samples / wave
#define SAMPLES_PER_WAVE 240    // < SEG, so a wave spans at most 2 segments

typedef float v2f __attribute__((ext_vector_type(2)));
typedef float v8f __attribute__((ext_vector_type(8)));

// ---------------------------------------------------------------------------
// Kernel 1: per-batch frame tables.
//   g[i] = max(f0[i], 20)
//   Q[i] = sum of f0_up over all samples t < (480 + 960*i)   (i = 0..255)
// Q[0] = 480*g[0] (head); each interior segment adds 480*(g[k]+g[k+1]).
// One block per batch item, 256 threads, LDS Hillis-Steele scan.
// ---------------------------------------------------------------------------
__global__ __launch_bounds__(256) void hosc_prefix(const float* __restrict__ f0,
                                                   float* __restrict__ G,
                                                   float* __restrict__ Q) {
  __shared__ float sg[LFRAMES];
  __shared__ float sp[LFRAMES];
  const int n = blockIdx.x;
  const int i = threadIdx.x;

  float g = fmaxf(f0[n * LFRAMES + i], 20.0f);
  sg[i] = g;
  __syncthreads();

  float p = (i < LFRAMES - 1) ? 480.0f * (sg[i] + sg[i + 1]) : 0.0f;
  sp[i] = p;
  __syncthreads();

  for (int off = 1; off < LFRAMES; off <<= 1) {
    float v = (i >= off) ? sp[i - off] : 0.0f;
    __syncthreads();
    sp[i] += v;
    __syncthreads();
  }

  const float q0 = 480.0f * sg[0];
  const float Qi = (i == 0) ? q0 : (q0 + sp[i - 1]);
  G[n * LFRAMES + i] = g;
  Q[n * LFRAMES + i] = Qi;
}

// ---------------------------------------------------------------------------
// Kernel 2: synthesis. 8 waves/block; each wave owns 240 consecutive samples
// (15 tiles of 16) of one batch item. The frame window for interpolation over
// those 240 samples is {ibase .. ibase+2}, so:
//   A (16x4, M=harmonic, K=frame offset) = exp(amps[n, h, ibase+K])  [loaded ONCE]
//   B (4x16 per tile)                    = row kk -> (1-w), row kk+1 -> w
//   D = A x B via V_WMMA_F32_16X16X4_F32 -> amps_up tile (lane = sample,
//                                           VGPR row = harmonic)
// Phase is closed-form (no scan): interior  Q[i] + (j+1)g[i] + d(j+1)^2/1920,
// head (t+1)g[0]; tail folds into interior since clamped g makes d = 0.
// sin(2*pi*x) == v_sin_f32(v_fract_f32(x))  (hardware sin takes revolutions).
// All per-iteration case handling is branch-free (v_cndmask, no EXEC churn);
// harmonic halves fold with one ds_swizzle SWAPX16; NT stores stream out.
// ---------------------------------------------------------------------------
__global__ __launch_bounds__(256) void hosc_synth(const float* __restrict__ amps,
                                                  const float* __restrict__ G,
                                                  const float* __restrict__ Q,
                                                  float* __restrict__ out) {
  const int tid  = threadIdx.x;
  const int wave = tid >> 5;
  const int lane = tid & 31;
  const bool hi  = (lane >= 16);
  const int l16  = lane & 15;
  const int n    = blockIdx.y;
  const int grp  = blockIdx.x * 8 + wave;       // 0..1023 per batch item
  const int tg0  = grp * SAMPLES_PER_WAVE;      // group start sample

  // Wave-uniform segment base (force into SGPR -> scalar loads below).
  int ib;
  if (tg0 < HEAD)            ib = 0;
  else if (tg0 >= TAILSTART) ib = LFRAMES - 1;
  else                       ib = (tg0 - HEAD) / SEG;
  const int ibase = __builtin_amdgcn_readfirstlane(ib);

  const float* Gn = G + n * LFRAMES;
  const float* Qn = Q + n * LFRAMES;
  const float G0 = Gn[ibase];
  const float G1 = Gn[min(ibase + 1, LFRAMES - 1)];
  const float G2 = Gn[min(ibase + 2, LFRAMES - 1)];
  const float Q0 = Qn[ibase];
  const float Q1 = Qn[min(ibase + 1, LFRAMES - 1)];

  // ---- A matrix (16x4 f32), loaded once per wave.
  // Lanes 0-15 hold K={0,1}, lanes 16-31 hold K={2,3}; M = l16.
  const int kb = hi ? 2 : 0;
  const float* arow = amps + (size_t)(n * NHARM + l16) * LFRAMES;
  v2f a;
  a.x = __expf(arow[min(ibase + kb,     LFRAMES - 1)]);  // v_exp_f32
  a.y = __expf(arow[min(ibase + kb + 1, LFRAMES - 1)]);

  const float harm0 = (float)((hi ? 8 : 0) + 1);

  for (int it = 0; it < TILES_PER_WAVE; ++it) {
    const int t = tg0 + it * 16 + l16;
    const int u = t - HEAD;

    // ---- Branch-free per-sample phase + interp weight.
    const bool head = (u < 0);
    const int  uu   = head ? 0 : u;          // head shares interior index math
    const int  i    = uu / SEG;              // 0..255 (255 only in the tail)
    const int  j    = uu - i * SEG;
    const int  kk   = i - ibase;             // 0 or 1 within this wave's window
    const bool s1   = (kk != 0);

    const float gi  = s1 ? G1 : G0;
    const float gi1 = s1 ? G2 : G1;          // clamped -> d=0 in the tail
    const float Qi  = s1 ? Q1 : Q0;
    const float jp  = (float)(j + 1);

    const float base_int  = (Qi + jp * gi +
                             (gi1 - gi) * jp * jp * (1.0f / 1920.0f))
                            * (1.0f / 48000.0f);
    const float base_head = (float)(t + 1) * G0 * (1.0f / 48000.0f);
    const float base = head ? base_head : base_int;

    const bool  wzero = head | (i == LFRAMES - 1);   // clipped regions
    const float w  = wzero ? 0.0f : ((float)j + 0.5f) * (1.0f / 960.0f);
    const float w0 = 1.0f - w;

    // ---- B matrix (4x16): row kk -> w0, row kk+1 -> w; others 0.
    // Straight-line selects only (no EXEC divergence before the WMMA).
    const float bx_lo = s1 ? 0.0f : w0;      // row K=0
    const float by_lo = s1 ? w0   : w;       // row K=1
    const float bx_hi = s1 ? w    : 0.0f;    // row K=2
    v2f b;
    b.x = hi ? bx_hi : bx_lo;
    b.y = hi ? 0.0f  : by_lo;

    v8f c = {0.f, 0.f, 0.f, 0.f, 0.f, 0.f, 0.f, 0.f};
    v8f d = __builtin_amdgcn_wmma_f32_16x16x4_f32(
        /*neg_a=*/false, a, /*neg_b=*/false, b,
        /*c_mod=*/(short)0, c, /*reuse_a=*/false, /*reuse_b=*/false);
    // d[r] @ lane = amps_up[harmonic M = r + (hi?8:0)][sample N = l16]

    // ---- 8 harmonics per lane: sin(2*pi*base*mul) = v_sin(fract(base*mul)).
    float acc = 0.0f;
#pragma unroll
    for (int r = 0; r < 8; ++r) {
      const float ph = __builtin_amdgcn_fractf(base * (harm0 + (float)r));
      acc = __builtin_fmaf(d[r], __builtin_amdgcn_sinf(ph), acc);
    }

    // Fold harmonic halves: lane L <-> L^16 (ds_swizzle SWAPX16: 0x401F).
    acc += __int_as_float(
        __builtin_amdgcn_ds_swizzle(__float_as_int(acc), 0x401F));

    if (!hi) {
      __builtin_nontemporal_store(acc * (1.0f / 16.0f),
                                  out + (size_t)n * LW + t);
    }
  }
}

// ---------------------------------------------------------------------------
extern "C" void kernel_launch(void* const* d_in, const int* in_sizes, int n_in,
                              void* d_out, int out_size, void* d_ws, size_t ws_size,
                              hipStream_t stream) {
  (void)in_sizes; (void)n_in; (void)out_size; (void)ws_size;
  const float* amps = (const float*)d_in[0];  // [16,16,256] f32
  const float* f0   = (const float*)d_in[1];  // [16,1,256]  f32
  float* out = (float*)d_out;                 // [16,1,245760] f32

  float* G = (float*)d_ws;                    // 16*256 floats
  float* Q = G + NB * LFRAMES;                // 16*256 floats (32 KB total)

  hosc_prefix<<<NB, LFRAMES, 0, stream>>>(f0, G, Q);

  // 1024 groups of 240 samples per batch item; 8 waves (groups) per block.
  dim3 grid(LW / (SAMPLES_PER_WAVE * 8), NB); // (128, 16)
  hosc_synth<<<grid, 256, 0, stream>>>(amps, G, Q, out);
}